// BiCutLoss_25632364822589
// MI455X (gfx1250) — compile-verified
//
#include <hip/hip_runtime.h>
#include <stdint.h>

// BiCutLoss (metric='nci'): memory-bound masked reduction.
// B rows x L=1024 cols, output[B,L,2] f32, labels[B,L] i32 -> scalar f32.
// HBM traffic ~192 MB -> ~8us at 23.3 TB/s; design = pure streaming.
// CDNA5 path used: async global->LDS DMA (global_load_async_to_lds_b128,
// s_wait_asynccnt) to stage each row while label loads overlap.

#define BICUT_ALPHA 0.65f
#define BICUT_L     1024
#define BICUT_TPB   256

__global__ __launch_bounds__(BICUT_TPB)
void bicut_row_kernel(const float* __restrict__ out,
                      const int*   __restrict__ labels,
                      float*       __restrict__ rowsums)
{
    __shared__ float2 s_out[BICUT_L];     // 8 KB: staged row, both channels
    __shared__ int    s_idx;
    __shared__ float  s_red[BICUT_TPB];

    const int row = blockIdx.x;
    const int t   = threadIdx.x;
    if (t == 0) s_idx = -1;

    // ---- async global -> LDS staging of this row (8 KB, b128 per lane) ----
    const uint32_t lds_base = (uint32_t)(size_t)(&s_out[0]);
    const uint64_t g_base   = (uint64_t)(size_t)(out + (size_t)row * (2u * BICUT_L));
#pragma unroll
    for (int k = 0; k < 2; ++k) {
        uint32_t boff = (uint32_t)(t + BICUT_TPB * k) * 16u;   // 16 B per lane
        uint64_t ga   = g_base + boff;
        uint32_t la   = lds_base + boff;
        asm volatile("global_load_async_to_lds_b128 %0, %1, off"
                     :: "v"(la), "v"(ga) : "memory");
    }

    // Overlap the DMA with the (coalesced) label loads for our 4 columns.
    int lab[4];
    const int* lrow = labels + (size_t)row * BICUT_L;
#pragma unroll
    for (int k = 0; k < 4; ++k) lab[k] = lrow[t + BICUT_TPB * k];

    asm volatile("s_wait_asynccnt 0x0" ::: "memory");  // own wave's DMA done
    __syncthreads();                                    // all waves' LDS visible

    // Pass 1: per-element decision + contribution (kept in registers).
    float c[4];
    int   lastz = -1;
#pragma unroll
    for (int k = 0; k < 4; ++k) {
        int    j = t + BICUT_TPB * k;
        float2 o = s_out[j];
        // temp = argmax([ch0, ch1]) (first-occurrence tie-break):
        //   temp==0  <=>  !(ch1 > ch0)
        if (!(o.y > o.x)) lastz = j;   // j ascending -> ends at last zero
        float jf = (float)j;
        float r1 = (lab[k] == 1) ? (-1.0f / __log2f(jf + 2.0f))
                                 : ((jf + 1.0f) / BICUT_ALPHA);
        c[k] = o.y * r1;
    }
    atomicMax(&s_idx, lastz);          // LDS ds_max, cheap
    __syncthreads();
    const int idx = (s_idx < 0) ? BICUT_L : s_idx;   // all-ones => idx = L

    // Pass 2: masked accumulation (mask: j <= idx).
    float partial = 0.0f;
#pragma unroll
    for (int k = 0; k < 4; ++k) {
        int j = t + BICUT_TPB * k;
        partial += (j <= idx) ? c[k] : 0.0f;
    }

    // Deterministic block tree reduction.
    s_red[t] = partial;
    __syncthreads();
#pragma unroll
    for (int off = BICUT_TPB / 2; off > 0; off >>= 1) {
        if (t < off) s_red[t] += s_red[t + off];
        __syncthreads();
    }
    if (t == 0) rowsums[row] = s_red[0];
}

__global__ __launch_bounds__(BICUT_TPB)
void bicut_reduce_kernel(const float* __restrict__ rowsums,
                         float* __restrict__ result, int B)
{
    __shared__ float s[BICUT_TPB];
    float acc = 0.0f;
    for (int i = threadIdx.x; i < B; i += BICUT_TPB)  // fixed order per thread
        acc += rowsums[i];
    s[threadIdx.x] = acc;
    __syncthreads();
#pragma unroll
    for (int off = BICUT_TPB / 2; off > 0; off >>= 1) {
        if (threadIdx.x < off) s[threadIdx.x] += s[threadIdx.x + off];
        __syncthreads();
    }
    if (threadIdx.x == 0) result[0] = s[0] / (float)B;
}

extern "C" void kernel_launch(void* const* d_in, const int* in_sizes, int n_in,
                              void* d_out, int out_size, void* d_ws, size_t ws_size,
                              hipStream_t stream)
{
    const float* out_p  = (const float*)d_in[0];   // [B, L, 2] f32
    const int*   labels = (const int*)d_in[1];     // [B, L] int
    const int B = in_sizes[1] / BICUT_L;

    float* rowsums = (float*)d_ws;                 // B floats of scratch

    bicut_row_kernel<<<B, BICUT_TPB, 0, stream>>>(out_p, labels, rowsums);
    bicut_reduce_kernel<<<1, BICUT_TPB, 0, stream>>>(rowsums, (float*)d_out, B);
}